// JreCell_71519795413680
// MI455X (gfx1250) — compile-verified
//
#include <hip/hip_runtime.h>
#include <math.h>

typedef __attribute__((ext_vector_type(2))) float v2f;
typedef __attribute__((ext_vector_type(8))) float v8f;

#define TT 768
#define NN 384
#define DD 256
#define FDIM 512

// ---------------- block reduction helpers (wave32) ----------------
__device__ __forceinline__ float block_reduce_sum(float v, float* red) {
    for (int off = 16; off > 0; off >>= 1) v += __shfl_down(v, off, 32);
    int wave = threadIdx.x >> 5, lane = threadIdx.x & 31;
    if (lane == 0) red[wave] = v;
    __syncthreads();
    if (threadIdx.x == 0) {
        float s = red[0];
        int nw = blockDim.x >> 5;
        for (int i = 1; i < nw; ++i) s += red[i];
        red[0] = s;
    }
    __syncthreads();
    float r = red[0];
    __syncthreads();
    return r;
}

__device__ __forceinline__ float block_reduce_max(float v, float* red) {
    for (int off = 16; off > 0; off >>= 1) v = fmaxf(v, __shfl_down(v, off, 32));
    int wave = threadIdx.x >> 5, lane = threadIdx.x & 31;
    if (lane == 0) red[wave] = v;
    __syncthreads();
    if (threadIdx.x == 0) {
        float s = red[0];
        int nw = blockDim.x >> 5;
        for (int i = 1; i < nw; ++i) s = fmaxf(s, red[i]);
        red[0] = s;
    }
    __syncthreads();
    float r = red[0];
    __syncthreads();
    return r;
}

// ---------------- WMMA fp32 GEMM:  C = A(MxK) * B(NxK)^T  [+beta*C][+bias][relu] ----
// One wave computes a 32x32 tile as 2x2 fragments (4 accumulators).
// V_WMMA_F32_16X16X4_F32 layouts:
//   A lane l: row m=l%16, kbase=(l>>4)*2 -> float2 {A[m,kb],A[m,kb+1]}
//   B lane l: col n=l%16, kbase=(l>>4)*2 -> float2 {B[n,kb],B[n,kb+1]}  (B row-major NxK)
//   D lane l, vgpr j: m = j + 8*(l>>4), n = l%16
// 2x2 blocking: 4 b64 loads feed 4 WMMAs per k-step (vs 8 loads unblocked),
// and the 4 independent accumulator chains hide WMMA latency without NOPs.
__global__ void gemm_nt_wmma(const float* __restrict__ A, int lda,
                             const float* __restrict__ B, int ldb,
                             float* __restrict__ C, int ldc,
                             int M, int N, int K,
                             const float* __restrict__ bias, int relu, int beta) {
    int wave = threadIdx.x >> 5;
    int lane = threadIdx.x & 31;
    int tilesN = N >> 5;
    int totalTiles = (M >> 5) * tilesN;
    int tile = blockIdx.x * (blockDim.x >> 5) + wave;
    if (tile >= totalTiles) return;          // wave-uniform: EXEC stays all-ones
    int tm = (tile / tilesN) << 5;
    int tn = (tile % tilesN) << 5;
    int r  = lane & 15;
    int kb = (lane >> 4) << 1;
    const float* Ap0 = A + (size_t)(tm + r) * lda + kb;
    const float* Ap1 = Ap0 + (size_t)16 * lda;
    const float* Bp0 = B + (size_t)(tn + r) * ldb + kb;
    const float* Bp1 = Bp0 + (size_t)16 * ldb;
    v8f c00 = {}, c01 = {}, c10 = {}, c11 = {};
#pragma unroll 2
    for (int k = 0; k < K; k += 4) {
        v2f a0 = *(const v2f*)(Ap0 + k);
        v2f a1 = *(const v2f*)(Ap1 + k);
        v2f b0 = *(const v2f*)(Bp0 + k);
        v2f b1 = *(const v2f*)(Bp1 + k);
        c00 = __builtin_amdgcn_wmma_f32_16x16x4_f32(false, a0, false, b0, (short)0, c00, false, false);
        c01 = __builtin_amdgcn_wmma_f32_16x16x4_f32(false, a0, false, b1, (short)0, c01, false, false);
        c10 = __builtin_amdgcn_wmma_f32_16x16x4_f32(false, a1, false, b0, (short)0, c10, false, false);
        c11 = __builtin_amdgcn_wmma_f32_16x16x4_f32(false, a1, false, b1, (short)0, c11, false, false);
    }
    int lgrp = lane >> 4;
    v8f accs[2][2] = {{c00, c01}, {c10, c11}};
#pragma unroll
    for (int mi = 0; mi < 2; ++mi) {
#pragma unroll
        for (int ni = 0; ni < 2; ++ni) {
            int n = tn + (ni << 4) + r;
            float bv = bias ? bias[n] : 0.0f;
            int mbase = tm + (mi << 4) + (lgrp << 3);
#pragma unroll
            for (int j = 0; j < 8; ++j) {
                int m = mbase + j;
                float val = accs[mi][ni][j] + bv;
                if (beta) val += C[(size_t)m * ldc + n];
                if (relu) val = fmaxf(val, 0.0f);
                C[(size_t)m * ldc + n] = val;
            }
        }
    }
}

// ---------------- small prep kernels ----------------
__global__ void pad_kernel(const float* __restrict__ vf, const float* __restrict__ vpe,
                           float* __restrict__ xpad) {
    int i = blockIdx.x * blockDim.x + threadIdx.x;
    int total = (TT + 2) * DD;
    if (i >= total) return;
    int row = i / DD;
    xpad[i] = (row == 0 || row == TT + 1) ? 0.0f : vf[i - DD] + vpe[i - DD];
}

__global__ void repack_kernel(const float* __restrict__ cw, float* __restrict__ wk) {
    int i = blockIdx.x * blockDim.x + threadIdx.x;
    if (i >= DD * DD * 3) return;
    int k = i % 3, oi = i / 3;
    wk[k * DD * DD + oi] = cw[i];   // wk_k[o*D+i] = conv_w[o,i,k]
}

__global__ void tn_kernel(const float* __restrict__ text, const float* __restrict__ tpe,
                          float* __restrict__ tn) {
    __shared__ float red[8];
    int n = blockIdx.x, d = threadIdx.x;
    float val = text[n * DD + d] + tpe[n * DD + d];
    float ss = block_reduce_sum(val * val, red);
    float norm = fmaxf(sqrtf(ss), 1e-12f);
    tn[n * DD + d] = val / norm;
}

__global__ void uv_kernel(const float* __restrict__ wt, const float* __restrict__ wv,
                          const float* __restrict__ Wi, float* __restrict__ u) {
    int f = blockIdx.x * blockDim.x + threadIdx.x;
    if (f >= FDIM) return;
    float st = 0.0f, sv = 0.0f;
    for (int o = 0; o < DD; ++o) {
        float w = Wi[o * FDIM + f];
        st = fmaf(wt[o], w, st);
        sv = fmaf(wv[o], w, sv);
    }
    u[f] = st;
    u[FDIM + f] = sv;
}

__global__ void rowmean_sig_kernel(const float* __restrict__ edges, float* __restrict__ wts) {
    __shared__ float red[8];
    int n = blockIdx.x, tid = threadIdx.x;
    float s = 0.0f;
    for (int j = tid; j < NN; j += blockDim.x) s += edges[n * NN + j];
    s = block_reduce_sum(s, red);
    if (tid == 0) wts[n] = 1.0f / (1.0f + expf(-s / (float)NN));
}

__global__ void scale_kernel(float* __restrict__ t2, const float* __restrict__ wts) {
    int i = blockIdx.x * blockDim.x + threadIdx.x;
    if (i >= NN * DD) return;
    t2[i] *= wts[i / DD];
}

// ---------------- relu-score kernel: st/sv[t,n] = sum_f u*relu(f1[t,f]+f2[n,f]) ----
__global__ void score_kernel(const float* __restrict__ f1, const float* __restrict__ f2,
                             const float* __restrict__ u,
                             float* __restrict__ st, float* __restrict__ sv) {
    __shared__ float s1[16][260];
    __shared__ float s2[16][260];
    __shared__ float sut[256];
    __shared__ float suv[256];
    int t0 = blockIdx.x << 4, n0 = blockIdx.y << 4;
    int tid = threadIdx.x;
    int tt = tid >> 4, nn = tid & 15;
    float at = 0.0f, av = 0.0f;
    for (int ch = 0; ch < FDIM; ch += 256) {
        __syncthreads();
        for (int i = tid; i < 16 * 256; i += 256) {
            int r = i >> 8, c = i & 255;
            s1[r][c] = f1[(size_t)(t0 + r) * FDIM + ch + c];
            s2[r][c] = f2[(size_t)(n0 + r) * FDIM + ch + c];
        }
        sut[tid] = u[ch + tid];
        suv[tid] = u[FDIM + ch + tid];
        __syncthreads();
#pragma unroll 4
        for (int f = 0; f < 256; ++f) {
            float x = fmaxf(s1[tt][f] + s2[nn][f], 0.0f);
            at = fmaf(sut[f], x, at);
            av = fmaf(suv[f], x, av);
        }
    }
    st[(size_t)(t0 + tt) * NN + n0 + nn] = at;
    sv[(size_t)(t0 + tt) * NN + n0 + nn] = av;
}

// ---------------- softmax over N + text_context + v_out ----------------
__global__ void ctx_t_kernel(const float* __restrict__ st, const float* __restrict__ t2,
                             const float* __restrict__ v, const float* __restrict__ in_video,
                             float* __restrict__ vout) {
    __shared__ float satt[NN];
    __shared__ float red[8];
    int t = blockIdx.x, tid = threadIdx.x;
    float m = -3.4e38f;
    for (int n = tid; n < NN; n += blockDim.x) {
        float s = st[(size_t)t * NN + n];
        satt[n] = s;
        m = fmaxf(m, s);
    }
    m = block_reduce_max(m, red);
    float sum = 0.0f;
    for (int n = tid; n < NN; n += blockDim.x) {
        float e = expf(satt[n] - m);
        satt[n] = e;
        sum += e;
    }
    sum = block_reduce_sum(sum, red);  // also syncs satt
    float inv = 1.0f / sum;
    int d = tid;  // blockDim == DD
    float ctx = 0.0f;
    for (int n = 0; n < NN; ++n) ctx = fmaf(satt[n], t2[(size_t)n * DD + d], ctx);
    ctx *= inv;
    vout[(size_t)t * DD + d] = 0.5f * ctx + 0.5f * v[(size_t)t * DD + d] + in_video[(size_t)t * DD + d];
}

// ---------------- softmax over T + video_context + t_out ----------------
__global__ void ctx_v_kernel(const float* __restrict__ sv, const float* __restrict__ v,
                             const float* __restrict__ t2, const float* __restrict__ in_text,
                             float* __restrict__ tout) {
    __shared__ float satt[TT];
    __shared__ float red[8];
    int n = blockIdx.x, tid = threadIdx.x;
    float m = -3.4e38f;
    for (int t = tid; t < TT; t += blockDim.x) {
        float s = sv[(size_t)t * NN + n];
        satt[t] = s;
        m = fmaxf(m, s);
    }
    m = block_reduce_max(m, red);
    float sum = 0.0f;
    for (int t = tid; t < TT; t += blockDim.x) {
        float e = expf(satt[t] - m);
        satt[t] = e;
        sum += e;
    }
    sum = block_reduce_sum(sum, red);
    float inv = 1.0f / sum;
    int d = tid;  // blockDim == DD
    float ctx = 0.0f;
    for (int t = 0; t < TT; ++t) ctx = fmaf(satt[t], v[(size_t)t * DD + d], ctx);
    ctx *= inv;
    tout[(size_t)n * DD + d] = 0.5f * ctx + 0.5f * t2[(size_t)n * DD + d] + in_text[(size_t)n * DD + d];
}

// ---------------- host launch ----------------
extern "C" void kernel_launch(void* const* d_in, const int* in_sizes, int n_in,
                              void* d_out, int out_size, void* d_ws, size_t ws_size,
                              hipStream_t stream) {
    (void)in_sizes; (void)n_in; (void)out_size; (void)ws_size;
    const float* video = (const float*)d_in[0];
    const float* text  = (const float*)d_in[1];
    const float* vpe   = (const float*)d_in[2];
    const float* tpe   = (const float*)d_in[3];
    const float* convw = (const float*)d_in[4];
    const float* convb = (const float*)d_in[5];
    const float* bilw  = (const float*)d_in[6];
    const float* linw  = (const float*)d_in[7];
    const float* linb  = (const float*)d_in[8];
    const float* W1    = (const float*)d_in[9];
    const float* b1    = (const float*)d_in[10];
    const float* W2    = (const float*)d_in[11];
    const float* Wi    = (const float*)d_in[12];
    // d_in[13] = bi, d_in[15] = bt, d_in[17] = bv : constants cancel in softmax
    const float* wt    = (const float*)d_in[14];
    const float* wv    = (const float*)d_in[16];
    float* out = (float*)d_out;

    float* ws = (float*)d_ws;
    size_t off = 0;
    float* xpad  = ws + off; off += (size_t)(TT + 2) * DD;   // 197120
    float* wk    = ws + off; off += (size_t)3 * DD * DD;     // 196608
    float* v     = ws + off; off += (size_t)TT * DD;         // 196608
    float* tn    = ws + off; off += (size_t)NN * DD;         // 98304
    float* tmp   = ws + off; off += (size_t)NN * DD;         // 98304
    float* edges = ws + off; off += (size_t)NN * NN;         // 147456
    float* wts   = ws + off; off += NN;                      // 384
    float* t2    = ws + off; off += (size_t)NN * DD;         // 98304
    float* f1    = ws + off; off += (size_t)TT * FDIM;       // 393216
    float* f2    = ws + off; off += (size_t)NN * FDIM;       // 196608
    float* u     = ws + off; off += 2 * FDIM;                // 1024
    float* st    = ws + off; off += (size_t)TT * NN;         // 294912
    float* sv    = ws + off; off += (size_t)TT * NN;         // 294912

    auto gemm = [&](const float* A, int lda, const float* B, int ldb, float* C, int ldc,
                    int M, int N, int K, const float* bias, int relu, int beta) {
        int tiles = (M / 32) * (N / 32);
        int blocks = (tiles + 7) / 8;
        gemm_nt_wmma<<<blocks, 256, 0, stream>>>(A, lda, B, ldb, C, ldc, M, N, K, bias, relu, beta);
    };

    // prep
    pad_kernel<<<((TT + 2) * DD + 255) / 256, 256, 0, stream>>>(video, vpe, xpad);
    repack_kernel<<<(DD * DD * 3 + 255) / 256, 256, 0, stream>>>(convw, wk);
    tn_kernel<<<NN, DD, 0, stream>>>(text, tpe, tn);
    uv_kernel<<<FDIM / 256, 256, 0, stream>>>(wt, wv, Wi, u);

    // VideoEncoder: conv1d(k=3,pad=1) as 3 accumulating GEMMs, bias+relu on last
    gemm(xpad + 0 * DD, DD, wk + 0 * DD * DD, DD, v, DD, TT, DD, DD, nullptr, 0, 0);
    gemm(xpad + 1 * DD, DD, wk + 1 * DD * DD, DD, v, DD, TT, DD, DD, nullptr, 0, 1);
    gemm(xpad + 2 * DD, DD, wk + 2 * DD * DD, DD, v, DD, TT, DD, DD, convb, 1, 1);

    // TextGnnEncoder
    gemm(tn, DD, bilw, DD, tmp, DD, NN, DD, DD, nullptr, 0, 0);     // tn @ bil_w^T
    gemm(tmp, DD, tn, DD, edges, NN, NN, NN, DD, nullptr, 0, 0);    // @ tn^T
    rowmean_sig_kernel<<<NN, 256, 0, stream>>>(edges, wts);
    gemm(tn, DD, linw, DD, t2, DD, NN, DD, DD, linb, 0, 0);         // tn @ lin_w^T + b
    scale_kernel<<<(NN * DD + 255) / 256, 256, 0, stream>>>(t2, wts);

    // PairInteraction projections
    gemm(v, DD, W1, DD, f1, FDIM, TT, FDIM, DD, b1, 0, 0);          // v @ W1^T + b1
    gemm(t2, DD, W2, DD, f2, FDIM, NN, FDIM, DD, nullptr, 0, 0);    // t2 @ W2^T

    // collapsed (t,n) attention scores (imap folded through wt/wv; consts cancel in softmax)
    score_kernel<<<dim3(TT / 16, NN / 16), 256, 0, stream>>>(f1, f2, u, st, sv);

    // dual softmax pooling + residual outputs
    ctx_t_kernel<<<TT, DD, 0, stream>>>(st, t2, v, video, out);            // v_out
    ctx_v_kernel<<<NN, DD, 0, stream>>>(sv, v, t2, text, out + TT * DD);   // t_out
}